// GATEncoder_79096117723246
// MI455X (gfx1250) — compile-verified
//
#include <hip/hip_runtime.h>
#include <hip/hip_bf16.h>

// ---------------------------------------------------------------------------
// GAT encoder (2x GATConv) for MI455X / gfx1250.
// Dense GEMMs use V_WMMA_F32_16X16X4_F32 (fp32 in/out, wave32), K templated
// for full unroll (32 straight-line WMMAs per 16x16 tile).
// Edge phase: uint-keyed atomic segment-max, atomic f32 segment-sum,
// atomic f32 weighted scatter-add. Everything lives in L2 (192 MB).
// ---------------------------------------------------------------------------

typedef float v2f __attribute__((ext_vector_type(2)));
typedef float v8f __attribute__((ext_vector_type(8)));

#define NEG_SLOPE 0.2f
// order-preserving uint encoding of float; key(-inf):
#define KEY_NEG_INF 0x007FFFFFu

__device__ __forceinline__ unsigned fkey(float x) {
  unsigned b = __float_as_uint(x);
  return (b & 0x80000000u) ? ~b : (b | 0x80000000u);
}
__device__ __forceinline__ float funkey(unsigned k) {
  return __uint_as_float((k & 0x80000000u) ? (k & 0x7FFFFFFFu) : ~k);
}

// ---------------------------- init kernels ---------------------------------

__global__ void fill_f32_kernel(float* __restrict__ p, int n, float v) {
  int i = blockIdx.x * blockDim.x + threadIdx.x;
  if (i < n) p[i] = v;
}
__global__ void fill_u32_kernel(unsigned* __restrict__ p, int n, unsigned v) {
  int i = blockIdx.x * blockDim.x + threadIdx.x;
  if (i < n) p[i] = v;
}

// ------------------------- WMMA fp32 GEMM ----------------------------------
// H[n][o] = sum_k A[n][k] * W[o][k]   (A:[N,K] row-major, W:[Cout,K] row-major)
// One wave (32 lanes) per 16x16 output tile. blockDim = (32, Cout/16).
// grid.x = N/16 (N divisible by 16). K compile-time, divisible by 4.
// A 16x4 layout: lane l holds M=l&15, K = k0 + 2*(l>>4) + {0,1}  (v2f)
// B 4x16 layout: lane l holds N=l&15, K = k0 + 2*(l>>4) + {0,1}  (v2f)
// D 16x16: lane l, vgpr v -> M = v + 8*(l>>4), N = l&15.

template <int K>
__global__ __launch_bounds__(256) void gemm_wmma_f32(
    const float* __restrict__ A, const float* __restrict__ W,
    float* __restrict__ H, int Cout) {
  const int lane = threadIdx.x;          // 0..31
  const int colTile = threadIdx.y;
  const int rowTile = blockIdx.x;

  const int mn    = lane & 15;
  const int khalf = (lane >> 4) * 2;     // 0 or 2

  const float* __restrict__ arow = A + (size_t)(rowTile * 16 + mn) * K + khalf;
  const float* __restrict__ wrow = W + (size_t)(colTile * 16 + mn) * K + khalf;

  v8f acc = {};
#pragma unroll
  for (int k0 = 0; k0 < K; k0 += 4) {
    v2f a = *(const v2f*)(arow + k0);
    v2f b = *(const v2f*)(wrow + k0);
    acc = __builtin_amdgcn_wmma_f32_16x16x4_f32(
        /*neg_a=*/false, a, /*neg_b=*/false, b,
        /*c_mod=*/(short)0, acc, /*reuse_a=*/false, /*reuse_b=*/false);
  }

  const int ncol = colTile * 16 + mn;
  const int mrow = rowTile * 16 + 8 * (lane >> 4);
#pragma unroll
  for (int v = 0; v < 8; ++v)
    H[(size_t)(mrow + v) * Cout + ncol] = acc[v];
}

// --------------------- per-node attention dots -----------------------------
// alpha_s[node*heads+h] = dot(H[node, h*C : (h+1)*C], att_src[h]); same for dst.

__global__ void alpha_kernel(const float* __restrict__ H,
                             const float* __restrict__ att_src,
                             const float* __restrict__ att_dst,
                             float* __restrict__ as_, float* __restrict__ ad_,
                             int n, int heads, int C) {
  int idx = blockIdx.x * blockDim.x + threadIdx.x;
  if (idx >= n * heads) return;
  int node = idx / heads, h = idx % heads;
  const float* hp = H + (size_t)node * heads * C + (size_t)h * C;
  const float* sp = att_src + h * C;
  const float* dp = att_dst + h * C;
  float ss = 0.f, sd = 0.f;
  for (int c = 0; c < C; c += 4) {
    float4 hv = *(const float4*)(hp + c);
    float4 sv = *(const float4*)(sp + c);
    float4 dv = *(const float4*)(dp + c);
    ss += hv.x * sv.x + hv.y * sv.y + hv.z * sv.z + hv.w * sv.w;
    sd += hv.x * dv.x + hv.y * dv.y + hv.z * dv.z + hv.w * dv.w;
  }
  as_[idx] = ss;
  ad_[idx] = sd;
}

// --------------------------- edge kernels ----------------------------------
// Edges 0..E-1 from edge_index; edges E..E+n-1 are self loops.

__device__ __forceinline__ void edge_endpoints(const int* __restrict__ ei,
                                               int E, int i, int& s, int& d) {
  if (i < E) { s = ei[i]; d = ei[E + i]; }
  else       { s = d = i - E; }
}

__global__ void edge_max_kernel(const int* __restrict__ ei, int E, int n,
                                int heads, const float* __restrict__ as_,
                                const float* __restrict__ ad_,
                                unsigned* __restrict__ segmax) {
  int i = blockIdx.x * blockDim.x + threadIdx.x;
  if (i >= E + n) return;
  int s, d;
  edge_endpoints(ei, E, i, s, d);
  for (int h = 0; h < heads; ++h) {
    float e = as_[s * heads + h] + ad_[d * heads + h];
    e = (e > 0.f) ? e : NEG_SLOPE * e;
    atomicMax(&segmax[d * heads + h], fkey(e));
  }
}

__global__ void edge_expsum_kernel(const int* __restrict__ ei, int E, int n,
                                   int heads, const float* __restrict__ as_,
                                   const float* __restrict__ ad_,
                                   const unsigned* __restrict__ segmax,
                                   float* __restrict__ exbuf,
                                   float* __restrict__ denom) {
  int i = blockIdx.x * blockDim.x + threadIdx.x;
  if (i >= E + n) return;
  int s, d;
  edge_endpoints(ei, E, i, s, d);
  for (int h = 0; h < heads; ++h) {
    float e = as_[s * heads + h] + ad_[d * heads + h];
    e = (e > 0.f) ? e : NEG_SLOPE * e;
    float m  = funkey(segmax[d * heads + h]);
    float ex = __expf(e - m);
    exbuf[(size_t)i * heads + h] = ex;
    atomicAdd(&denom[d * heads + h], ex);
  }
}

// Weighted scatter-add: one thread per (edge, channel).
__global__ void aggregate_kernel(const int* __restrict__ ei, int E, int n,
                                 int heads, int C,
                                 const float* __restrict__ H,
                                 const float* __restrict__ exbuf,
                                 const float* __restrict__ denom,
                                 float* __restrict__ out) {
  int CH = heads * C;
  int idx = blockIdx.x * blockDim.x + threadIdx.x;
  int total = (E + n) * CH;   // <= 108.8M, fits int
  if (idx >= total) return;
  int i = idx / CH;
  int c = idx % CH;
  int h = c / C;
  int s, d;
  edge_endpoints(ei, E, i, s, d);
  float alpha = exbuf[(size_t)i * heads + h] /
                (denom[d * heads + h] + 1e-16f);
  atomicAdd(&out[(size_t)d * CH + c], H[(size_t)s * CH + c] * alpha);
}

// ------------------------ elementwise epilogues ----------------------------

__global__ void bias_elu_kernel(float* __restrict__ x,
                                const float* __restrict__ b, int n, int CH) {
  int idx = blockIdx.x * blockDim.x + threadIdx.x;
  if (idx >= n * CH) return;
  float y = x[idx] + b[idx % CH];
  x[idx] = (y > 0.f) ? y : expm1f(y);
}

__global__ void bias_add_kernel(float* __restrict__ x,
                                const float* __restrict__ b, int n, int CH) {
  int idx = blockIdx.x * blockDim.x + threadIdx.x;
  if (idx >= n * CH) return;
  x[idx] = x[idx] + b[idx % CH];
}

// ------------------------------ host side ----------------------------------

static inline int cdiv(long long a, long long b) { return (int)((a + b - 1) / b); }

extern "C" void kernel_launch(void* const* d_in, const int* in_sizes, int n_in,
                              void* d_out, int out_size, void* d_ws, size_t ws_size,
                              hipStream_t stream) {
  const int IN_CH = 128, HID = 64, HEADS = 2, OUT = 64;
  const int CH1 = HEADS * HID;  // 128

  const float* x        = (const float*)d_in[0];
  const int*   ei       = (const int*)d_in[1];
  const float* W1       = (const float*)d_in[2];
  const float* att1_src = (const float*)d_in[3];
  const float* att1_dst = (const float*)d_in[4];
  const float* b1       = (const float*)d_in[5];
  const float* W2       = (const float*)d_in[6];
  const float* att2_src = (const float*)d_in[7];
  const float* att2_dst = (const float*)d_in[8];
  const float* b2       = (const float*)d_in[9];
  float* out            = (float*)d_out;

  const int n  = in_sizes[0] / IN_CH;   // 50000
  const int E  = in_sizes[1] / 2;       // 800000
  const int Ep = E + n;                 // edges incl. self-loops

  // ---- workspace bump allocator ----
  char* ws = (char*)d_ws;
  size_t off = 0;
  auto alloc = [&](size_t bytes) -> void* {
    void* p = ws + off;
    off += (bytes + 255) & ~(size_t)255;
    return p;
  };
  float*    h1      = (float*)   alloc((size_t)n * CH1 * 4);
  float*    agg1    = (float*)   alloc((size_t)n * CH1 * 4);  // -> layer2 input
  float*    h2      = (float*)   alloc((size_t)n * OUT * 4);
  float*    as1     = (float*)   alloc((size_t)n * HEADS * 4);
  float*    ad1     = (float*)   alloc((size_t)n * HEADS * 4);
  unsigned* segmax1 = (unsigned*)alloc((size_t)n * HEADS * 4);
  float*    denom1  = (float*)   alloc((size_t)n * HEADS * 4);
  float*    as2     = (float*)   alloc((size_t)n * 4);
  float*    ad2     = (float*)   alloc((size_t)n * 4);
  unsigned* segmax2 = (unsigned*)alloc((size_t)n * 4);
  float*    denom2  = (float*)   alloc((size_t)n * 4);
  float*    ex1     = (float*)   alloc((size_t)Ep * HEADS * 4);
  float*    ex2     = (float*)   alloc((size_t)Ep * 4);
  (void)ws_size;

  const int BT = 256;

  // ---- init accumulators (harness poisons ws/out each run) ----
  fill_f32_kernel<<<cdiv((long long)n * CH1, BT), BT, 0, stream>>>(agg1, n * CH1, 0.f);
  fill_f32_kernel<<<cdiv(n * HEADS, BT), BT, 0, stream>>>(denom1, n * HEADS, 0.f);
  fill_u32_kernel<<<cdiv(n * HEADS, BT), BT, 0, stream>>>(segmax1, n * HEADS, KEY_NEG_INF);
  fill_f32_kernel<<<cdiv((long long)n * OUT, BT), BT, 0, stream>>>(out, n * OUT, 0.f);
  fill_f32_kernel<<<cdiv(n, BT), BT, 0, stream>>>(denom2, n, 0.f);
  fill_u32_kernel<<<cdiv(n, BT), BT, 0, stream>>>(segmax2, n, KEY_NEG_INF);

  // =================== layer 1 ===================
  // h1 = x @ W1^T   (N x 128, K=128)
  {
    dim3 block(32, CH1 / 16);  // 8 waves per block
    dim3 grid(n / 16);
    gemm_wmma_f32<128><<<grid, block, 0, stream>>>(x, W1, h1, CH1);
  }
  alpha_kernel<<<cdiv(n * HEADS, BT), BT, 0, stream>>>(
      h1, att1_src, att1_dst, as1, ad1, n, HEADS, HID);
  edge_max_kernel<<<cdiv(Ep, BT), BT, 0, stream>>>(ei, E, n, HEADS, as1, ad1, segmax1);
  edge_expsum_kernel<<<cdiv(Ep, BT), BT, 0, stream>>>(
      ei, E, n, HEADS, as1, ad1, segmax1, ex1, denom1);
  aggregate_kernel<<<cdiv((long long)Ep * CH1, BT), BT, 0, stream>>>(
      ei, E, n, HEADS, HID, h1, ex1, denom1, agg1);
  bias_elu_kernel<<<cdiv((long long)n * CH1, BT), BT, 0, stream>>>(agg1, b1, n, CH1);

  // =================== layer 2 ===================
  // h2 = agg1 @ W2^T  (N x 64, K=128)
  {
    dim3 block(32, OUT / 16);  // 4 waves per block
    dim3 grid(n / 16);
    gemm_wmma_f32<128><<<grid, block, 0, stream>>>(agg1, W2, h2, OUT);
  }
  alpha_kernel<<<cdiv(n, BT), BT, 0, stream>>>(
      h2, att2_src, att2_dst, as2, ad2, n, 1, OUT);
  edge_max_kernel<<<cdiv(Ep, BT), BT, 0, stream>>>(ei, E, n, 1, as2, ad2, segmax2);
  edge_expsum_kernel<<<cdiv(Ep, BT), BT, 0, stream>>>(
      ei, E, n, 1, as2, ad2, segmax2, ex2, denom2);
  aggregate_kernel<<<cdiv((long long)Ep * OUT, BT), BT, 0, stream>>>(
      ei, E, n, 1, OUT, h2, ex2, denom2, out);
  bias_add_kernel<<<cdiv((long long)n * OUT, BT), BT, 0, stream>>>(out, b2, n, OUT);
}